// IncrementalClassRectificationLoss_9294309228789
// MI455X (gfx1250) — compile-verified
//
#include <hip/hip_runtime.h>

#define BN 256      // batch
#define CN 28       // classes
#define CP 32       // padded classes (WMMA N tiles)
#define DN 2048     // embedding dim
#define KN 16       // top-k

typedef __attribute__((ext_vector_type(2))) float v2f;
typedef __attribute__((ext_vector_type(8))) float v8f;

// ---------------- Kernel A0: counts + minority mask + zero NegSel ----------
__global__ void crl_stats(const int* __restrict__ target, int* __restrict__ counts,
                          int* __restrict__ minority, float* __restrict__ NegSel) {
  int tid = threadIdx.x;
  for (int t = tid; t < BN * CP; t += blockDim.x) NegSel[t] = 0.0f;
  if (tid < CN) {
    int s = 0;
    for (int i = 0; i < BN; ++i) s += target[i * CN + tid];
    counts[tid] = s;
  }
  __syncthreads();
  if (tid == 0) {
    // stable ascending insertion sort of classes by count (matches jnp.argsort)
    int ord[CN];
    for (int c = 0; c < CN; ++c) ord[c] = c;
    for (int a = 1; a < CN; ++a) {
      int v = ord[a]; int b = a - 1;
      while (b >= 0 && counts[ord[b]] > counts[v]) { ord[b + 1] = ord[b]; --b; }
      ord[b + 1] = v;
    }
    float cum = 0.0f;
    for (int p = 0; p < CN; ++p) {
      cum += (float)counts[ord[p]];
      int keep = (cum <= 0.5f * (float)CN) ? 1 : 0;
      minority[ord[p]] = (keep && counts[ord[p]] > 1) ? 1 : 0;
    }
  }
}

// ---------------- Kernel A1: per-class top-K selections (1 wave32 / class) --
__global__ void crl_select(const float* __restrict__ inp, const int* __restrict__ target,
                           const int* __restrict__ counts, float* __restrict__ NegSel,
                           int* __restrict__ n_n, int* __restrict__ n_pe,
                           int* __restrict__ posCnt, int* __restrict__ posList) {
  const int c = blockIdx.x;
  const int lane = threadIdx.x;        // 0..31, wave32
  const float INF = __builtin_inff();
  float negv[8], posv[8];
  for (int t = 0; t < 8; ++t) {
    int r = lane + 32 * t;
    float p = inp[r * CN + c];
    int tg = target[r * CN + c];
    negv[t] = tg ? -INF : p;           // negatives keyed by prob (largest = hardest)
    posv[t] = tg ? p : INF;            // positives keyed by prob (smallest = hardest)
  }
  // hardest negatives: K largest
  int nn = 0;
  for (int s = 0; s < KN; ++s) {
    float bv = -INF; int bi = 0x3fffffff;
    for (int t = 0; t < 8; ++t) {
      int r = lane + 32 * t;
      if (negv[t] > bv || (negv[t] == bv && r < bi)) { bv = negv[t]; bi = r; }
    }
    for (int off = 16; off >= 1; off >>= 1) {
      float ov = __shfl_xor(bv, off, 32);
      int   oi = __shfl_xor(bi, off, 32);
      if (ov > bv || (ov == bv && oi < bi)) { bv = ov; bi = oi; }
    }
    if (!(bv > -1e37f)) break;         // exhausted (uniform across wave)
    ++nn;
    if (lane == 0) NegSel[bi * CP + c] = 1.0f;
    if ((bi & 31) == lane) negv[bi >> 5] = -INF;
  }
  // hardest positives: K+1 smallest (so any anchor can be excluded exactly)
  int np = 0;
  for (int s = 0; s < KN + 1; ++s) {
    float bv = INF; int bi = 0x3fffffff;
    for (int t = 0; t < 8; ++t) {
      int r = lane + 32 * t;
      if (posv[t] < bv || (posv[t] == bv && r < bi)) { bv = posv[t]; bi = r; }
    }
    for (int off = 16; off >= 1; off >>= 1) {
      float ov = __shfl_xor(bv, off, 32);
      int   oi = __shfl_xor(bi, off, 32);
      if (ov < bv || (ov == bv && oi < bi)) { bv = ov; bi = oi; }
    }
    if (!(bv < 1e37f)) break;
    if (lane == 0) posList[c * (KN + 1) + np] = bi;
    ++np;
    if ((bi & 31) == lane) posv[bi >> 5] = INF;
  }
  if (lane == 0) {
    n_n[c] = nn;
    posCnt[c] = np;
    int pe = counts[c] - 1;
    if (pe > KN) pe = KN;
    if (pe < 0) pe = 0;
    n_pe[c] = pe;                      // n_p when the anchor is a positive
  }
}

// ---------------- Kernel B: pairwise L1 distance, 32x32 tile / workgroup ----
#define TK 64
__global__ void crl_l1dist(const float* __restrict__ X, float* __restrict__ Dm) {
  __shared__ float As[32][TK + 1];     // +1 pad: conflict-free column reads
  __shared__ float Bs[32][TK + 1];
  const int tid = threadIdx.x;
  const int tx = tid & 15, ty = tid >> 4;
  const int ib = blockIdx.y * 32, jb = blockIdx.x * 32;
  const int lr = tid >> 3;             // 0..31 row to stage
  const int lc = (tid & 7) * 8;        // 0..56 col base (8 floats / thread)
  float a00 = 0.f, a01 = 0.f, a10 = 0.f, a11 = 0.f;
  for (int kb = 0; kb < DN; kb += TK) {
    const float4* pa = (const float4*)(X + (ib + lr) * DN + kb + lc);
    const float4* pb = (const float4*)(X + (jb + lr) * DN + kb + lc);
    float4 va0 = pa[0], va1 = pa[1];
    float4 vb0 = pb[0], vb1 = pb[1];
    if (kb + TK < DN) {                // gfx1250 global_prefetch_b8 for next tile
      __builtin_prefetch(X + (ib + lr) * DN + kb + TK + lc, 0, 1);
      __builtin_prefetch(X + (jb + lr) * DN + kb + TK + lc, 0, 1);
    }
    __syncthreads();
    As[lr][lc + 0] = va0.x; As[lr][lc + 1] = va0.y; As[lr][lc + 2] = va0.z; As[lr][lc + 3] = va0.w;
    As[lr][lc + 4] = va1.x; As[lr][lc + 5] = va1.y; As[lr][lc + 6] = va1.z; As[lr][lc + 7] = va1.w;
    Bs[lr][lc + 0] = vb0.x; Bs[lr][lc + 1] = vb0.y; Bs[lr][lc + 2] = vb0.z; Bs[lr][lc + 3] = vb0.w;
    Bs[lr][lc + 4] = vb1.x; Bs[lr][lc + 5] = vb1.y; Bs[lr][lc + 6] = vb1.z; Bs[lr][lc + 7] = vb1.w;
    __syncthreads();
#pragma unroll 16
    for (int k = 0; k < TK; ++k) {
      float x0 = As[2 * ty][k],     x1 = As[2 * ty + 1][k];
      float y0 = Bs[2 * tx][k],     y1 = Bs[2 * tx + 1][k];
      a00 += __builtin_fabsf(x0 - y0);
      a01 += __builtin_fabsf(x0 - y1);
      a10 += __builtin_fabsf(x1 - y0);
      a11 += __builtin_fabsf(x1 - y1);
    }
  }
  const int i0 = ib + 2 * ty, j0 = jb + 2 * tx;
  Dm[i0 * BN + j0]           = a00;
  Dm[i0 * BN + j0 + 1]       = a01;
  Dm[(i0 + 1) * BN + j0]     = a10;
  Dm[(i0 + 1) * BN + j0 + 1] = a11;
}

// ---------------- Kernel C: DnAll = Dm(256x256) @ NegSel(256x32) via WMMA ---
__global__ void crl_dn_wmma(const float* __restrict__ Dm, const float* __restrict__ NegSel,
                            float* __restrict__ DnAll) {
  const int lane = threadIdx.x;        // one wave32 per block
  const int half = lane >> 4;          // lane half selects K pair (ISA f32 A/B layout)
  const int l16  = lane & 15;
  const int rowT = blockIdx.y * 16;
  const int colT = blockIdx.x * 16;
  v8f acc;
#pragma unroll
  for (int r = 0; r < 8; ++r) acc[r] = 0.0f;
#pragma unroll 4
  for (int k = 0; k < BN; k += 4) {
    const int ka = k + 2 * half;
    v2f a, b;
    // A 16x4 f32: VGPR0 = {K=k (lanes0-15), K=k+2 (lanes16-31)}, VGPR1 = {K=k+1, K=k+3}
    a.x = Dm[(rowT + l16) * BN + ka];
    a.y = Dm[(rowT + l16) * BN + ka + 1];
    // B 4x16 f32: rows striped across lanes within a VGPR, same half/pair scheme
    b.x = NegSel[ka * CP + colT + l16];
    b.y = NegSel[(ka + 1) * CP + colT + l16];
    acc = __builtin_amdgcn_wmma_f32_16x16x4_f32(false, a, false, b, (short)0, acc,
                                                false, false);
  }
#pragma unroll
  for (int r = 0; r < 8; ++r) {
    const int row = rowT + r + 8 * half;   // C/D: VGPR r holds M=r (lo lanes), M=r+8 (hi)
    DnAll[row * CP + colT + l16] = acc[r];
  }
}

// ---------------- Kernel D: weighted reduction + margin relu ----------------
__global__ void crl_final(const int* __restrict__ target, const float* __restrict__ Dm,
                          const float* __restrict__ DnAll, const int* __restrict__ minority,
                          const int* __restrict__ n_n, const int* __restrict__ n_pe,
                          const int* __restrict__ posCnt, const int* __restrict__ posList,
                          float* __restrict__ out) {
  __shared__ float sdp[256], sdn[256];
  const int i = threadIdx.x;           // anchor
  float ldp = 0.0f, ldn = 0.0f;
  for (int c = 0; c < CN; ++c) {
    if (target[i * CN + c] != 0 && minority[c]) {
      const int L = posCnt[c];
      float dp = 0.0f; int cnt = 0;
      for (int t = 0; t < L; ++t) {
        const int idx = posList[c * (KN + 1) + t];
        if (idx == i) continue;        // exclude anchor, keep first K of the rest
        if (cnt < KN) { dp += Dm[i * BN + idx]; ++cnt; }
      }
      ldp += (float)n_n[c] * dp;
      ldn += (float)n_pe[c] * DnAll[i * CP + c];
    }
  }
  sdp[i] = ldp; sdn[i] = ldn;
  __syncthreads();
  for (int s = 128; s > 0; s >>= 1) {
    if (i < s) { sdp[i] += sdp[i + s]; sdn[i] += sdn[i + s]; }
    __syncthreads();
  }
  if (i == 0) {
    float v = sdp[0] - sdn[0] + 1.0f;  // MARGIN
    out[0] = v > 0.0f ? v : 0.0f;
  }
}

extern "C" void kernel_launch(void* const* d_in, const int* in_sizes, int n_in,
                              void* d_out, int out_size, void* d_ws, size_t ws_size,
                              hipStream_t stream) {
  const float* inp    = (const float*)d_in[0];
  const int*   target = (const int*)d_in[1];
  const float* X      = (const float*)d_in[2];
  float* out = (float*)d_out;

  float* Dm      = (float*)d_ws;           // 256*256
  float* NegSel  = Dm + BN * BN;           // 256*32
  float* DnAll   = NegSel + BN * CP;       // 256*32
  int*   counts  = (int*)(DnAll + BN * CP);
  int*   minority = counts + CP;
  int*   nn_     = minority + CP;
  int*   npe_    = nn_ + CP;
  int*   posCnt  = npe_ + CP;
  int*   posList = posCnt + CP;            // 28*17

  crl_stats<<<1, 256, 0, stream>>>(target, counts, minority, NegSel);
  crl_select<<<CN, 32, 0, stream>>>(inp, target, counts, NegSel, nn_, npe_, posCnt, posList);
  crl_l1dist<<<dim3(8, 8), 256, 0, stream>>>(X, Dm);
  crl_dn_wmma<<<dim3(2, 16), 32, 0, stream>>>(Dm, NegSel, DnAll);
  crl_final<<<1, 256, 0, stream>>>(target, Dm, DnAll, minority, nn_, npe_, posCnt, posList, out);
}